// TopK_14680198218056
// MI455X (gfx1250) — compile-verified
//
#include <hip/hip_runtime.h>
#include <stdint.h>

// Problem constants (fixed by the reference)
#define NROWS 8192
#define NC    24576
#define KSEL  64
#define BLOCK 1024          // 32 waves (wave32)
#define NREP  16            // LDS histogram replicas (64 threads per replica)

#define SROW_BYTES  (NC * 4)             // 98304 B: the staged row
#define HIST_BYTES  (NREP * 256 * 4)     // 16384 B: replicated histograms
#define HSUM_BYTES  (256 * 4)            // 1024 B: merged histogram
#define SCAL_BYTES  (32)                 // control scalars
#define SMEM_BYTES  (SROW_BYTES + HIST_BYTES + HSUM_BYTES + SCAL_BYTES)  // ~113 KB

#if defined(__has_builtin)
# if __has_builtin(__builtin_amdgcn_global_load_async_to_lds_b128)
#  define HAVE_ASYNC_LDS 1
# endif
#endif

typedef float v4f __attribute__((ext_vector_type(4)));

// Pointer types matching the async-LDS builtin's expected signature:
// (int4 in global AS, int4 in LDS AS, imm offset, imm cpol)
typedef int v4i __attribute__((vector_size(4 * sizeof(int))));
typedef __attribute__((address_space(1))) v4i* gv4i_p;
typedef __attribute__((address_space(3))) v4i* lv4i_p;

// Monotone float -> uint key: larger float => larger key (no NaNs in input)
__device__ __forceinline__ uint32_t f2key(float f) {
  uint32_t u = __float_as_uint(f);
  return (u & 0x80000000u) ? ~u : (u | 0x80000000u);
}

__global__ __launch_bounds__(BLOCK, 1)
void topk_relu_scatter_kernel(const float* __restrict__ x, float* __restrict__ out) {
  extern __shared__ char smem[];
  float*    srow = (float*)smem;
  uint32_t* hist = (uint32_t*)(smem + SROW_BYTES);
  uint32_t* hsum = (uint32_t*)(smem + SROW_BYTES + HIST_BYTES);
  uint32_t* scal = (uint32_t*)(smem + SROW_BYTES + HIST_BYTES + HSUM_BYTES);
  // scal[0]=prefix value  scal[1]=prefix mask  scal[2]=need  scal[3]=E  scal[4]=threshold key

  const int tid = threadIdx.x;
  const int row = blockIdx.x;
  const float* __restrict__ grow = x   + (size_t)row * NC;
  float*       __restrict__ orow = out + (size_t)row * NC;

  // ---- Stage the full 96KB row into LDS via async global->LDS DMA ----
#if HAVE_ASYNC_LDS
  for (int v = tid; v < NC / 4; v += BLOCK) {
    __builtin_amdgcn_global_load_async_to_lds_b128(
        (gv4i_p)(grow + v * 4),
        (lv4i_p)(srow + v * 4),
        0, 0);
  }
#else
  for (int v = tid; v < NC / 4; v += BLOCK)
    ((v4f*)srow)[v] = ((const v4f*)grow)[v];
#endif

  if (tid == 0) { scal[0] = 0u; scal[1] = 0u; scal[2] = KSEL; scal[3] = 0u; scal[4] = 0u; }

#if HAVE_ASYNC_LDS
# if __has_builtin(__builtin_amdgcn_s_wait_asynccnt)
  __builtin_amdgcn_s_wait_asynccnt(0);
# else
  asm volatile("s_wait_asynccnt 0" ::: "memory");
# endif
#endif
  __syncthreads();

  const int myrep = tid >> 6;                 // 0..15
  const v4f* srow4 = (const v4f*)srow;

  // ---- 4-round radix select (8 bits/round) for the 64th-largest key ----
  for (int round = 0; round < 4; ++round) {
    const int shift = 24 - 8 * round;

    for (int i = tid; i < NREP * 256; i += BLOCK) hist[i] = 0u;
    __syncthreads();

    const uint32_t mask = scal[1];
    const uint32_t pref = scal[0];

    for (int v = tid; v < NC / 4; v += BLOCK) {
      v4f d = srow4[v];
#pragma unroll
      for (int c = 0; c < 4; ++c) {
        uint32_t k = f2key(d[c]);
        if ((k & mask) == pref)
          atomicAdd(&hist[myrep * 256 + ((k >> shift) & 255u)], 1u);
      }
    }
    __syncthreads();

    if (tid < 256) {
      uint32_t s = 0;
#pragma unroll
      for (int r = 0; r < NREP; ++r) s += hist[r * 256 + tid];
      hsum[tid] = s;
    }
    __syncthreads();

    if (tid == 0) {
      uint32_t need = scal[2], cum = 0;
      int b = 255;
      for (; b > 0; --b) {
        uint32_t c = hsum[b];
        if (cum + c >= need) break;
        cum += c;
      }
      scal[2] = need - cum;                       // equals still needed within bin b
      scal[0] = pref | ((uint32_t)b << shift);
      scal[1] = mask | (0xFFu << shift);
      if (round == 3) { scal[3] = hsum[b]; scal[4] = scal[0]; }
    }
    __syncthreads();
  }

  const uint32_t thresh = scal[4];
  const uint32_t need   = scal[2];   // how many threshold-equal elements belong in top-K
  const uint32_t E      = scal[3];   // total threshold-equal elements in the row
  const bool ranked = (E != need);   // only rank ties when duplicates exceed the quota

  // ---- Single output pass: relu(x) where selected, 0 elsewhere ----
  for (int v = tid; v < NC / 4; v += BLOCK) {
    v4f d = srow4[v];
    v4f o;
#pragma unroll
    for (int c = 0; c < 4; ++c) {
      uint32_t k = f2key(d[c]);
      float r = fmaxf(d[c], 0.0f);
      float oo = 0.0f;
      if (k > thresh) {
        oo = r;
      } else if (k == thresh) {
        if (!ranked) {
          oo = r;                               // all equals accepted (common case)
        } else {
          // Rare: duplicates at the threshold. JAX tie-break = lowest index first.
          int i = v * 4 + c;
          uint32_t rank = 0;
          for (int j = 0; j < i; ++j) rank += (f2key(srow[j]) == thresh) ? 1u : 0u;
          if (rank < need) oo = r;
        }
      }
      o[c] = oo;
    }
    __builtin_nontemporal_store(o, (v4f*)(orow + v * 4));
  }
}

extern "C" void kernel_launch(void* const* d_in, const int* in_sizes, int n_in,
                              void* d_out, int out_size, void* d_ws, size_t ws_size,
                              hipStream_t stream) {
  (void)in_sizes; (void)n_in; (void)out_size; (void)d_ws; (void)ws_size;
  const float* x  = (const float*)d_in[0];
  float*      out = (float*)d_out;
  hipLaunchKernelGGL(topk_relu_scatter_kernel, dim3(NROWS), dim3(BLOCK),
                     SMEM_BYTES, stream, x, out);
}